// InvariantPointAttention_4698694222445
// MI455X (gfx1250) — compile-verified
//
#include <hip/hip_runtime.h>
#include <hip/hip_bf16.h>

#define N_RES 1024
#define C1V   384
#define C2V   128
#define HV    12
#define SQKV  16
#define SVV   16
#define PQKV  4
#define PVV   8
#define OUTV  384
#define DOUTV 2112   // 192 + 288 + 96 + 1536

typedef float v2f __attribute__((ext_vector_type(2)));
typedef float v8f __attribute__((ext_vector_type(8)));

__device__ __forceinline__ v8f wmma_f32_k4(v2f a, v2f b, v8f c) {
  // D = A(16x4) * B(4x16) + C, fp32 WMMA (VOP3P V_WMMA_F32_16X16X4_F32)
  return __builtin_amdgcn_wmma_f32_16x16x4_f32(false, a, false, b, (short)0, c,
                                               false, false);
}

__device__ __forceinline__ int wave_linear_id() {
  return blockIdx.x * (blockDim.x >> 5) + (threadIdx.x >> 5);
}

// ---------------------------------------------------------------------------
// Generic fp32 WMMA GEMM: C[M,Nc] = A[M,K] @ W[K,Nc] + bias[Nc]
// One wave per 16x16 output tile. M%16==0, K%4==0, Nc%16==0.
// ---------------------------------------------------------------------------
__global__ void gemm_wmma_f32(const float* __restrict__ A,
                              const float* __restrict__ W,
                              const float* __restrict__ bias,
                              float* __restrict__ C,
                              int M, int K, int Nc) {
  const int tn = Nc >> 4;
  const int total = (M >> 4) * tn;
  int w = wave_linear_id();
  if (w >= total) return;
  int nt = w % tn, mt = w / tn;
  int row0 = mt << 4, col0 = nt << 4;
  int l = threadIdx.x & 31, half = l >> 4, lm = l & 15;

  v8f acc = {0.f, 0.f, 0.f, 0.f, 0.f, 0.f, 0.f, 0.f};
  const float* arow = A + (size_t)(row0 + lm) * K + 2 * half;
  const float* bcol = W + (size_t)2 * half * Nc + col0 + lm;
#pragma unroll 4
  for (int kk = 0; kk < K; kk += 4) {
    v2f a, b;
    a.x = arow[kk];
    a.y = arow[kk + 1];
    b.x = bcol[(size_t)kk * Nc];
    b.y = bcol[(size_t)(kk + 1) * Nc];
    acc = wmma_f32_k4(a, b, acc);
  }
  float bs = bias ? bias[col0 + lm] : 0.f;
#pragma unroll
  for (int r = 0; r < 8; ++r) {
    int m = row0 + r + 8 * half;
    C[(size_t)m * Nc + col0 + lm] = acc[r] + bs;
  }
}

// ---------------------------------------------------------------------------
// Apply per-residue rigid frame (rot, trans) to projected points;
// build q_pt/k_pt/v_pt in [c][h][n][p] layout and sq/sk = sum of squares.
// ---------------------------------------------------------------------------
__global__ void rotate_points_kernel(const float* __restrict__ qpl,
                                     const float* __restrict__ kvpl,
                                     const float* __restrict__ rot,
                                     const float* __restrict__ trans,
                                     float* __restrict__ qpt,
                                     float* __restrict__ kpt,
                                     float* __restrict__ vpt,
                                     float* __restrict__ sq,
                                     float* __restrict__ sk) {
  int n = blockIdx.x * blockDim.x + threadIdx.x;
  if (n >= N_RES) return;
  float R[9], t[3];
#pragma unroll
  for (int i = 0; i < 9; ++i) R[i] = rot[n * 9 + i];
#pragma unroll
  for (int i = 0; i < 3; ++i) t[i] = trans[n * 3 + i];

  float sqh[HV], skh[HV];
#pragma unroll
  for (int h = 0; h < HV; ++h) { sqh[h] = 0.f; skh[h] = 0.f; }

  // query points: qpl is [n][3][H*PQK]
  for (int k = 0; k < HV * PQKV; ++k) {
    float x0 = qpl[n * 144 + k];
    float x1 = qpl[n * 144 + 48 + k];
    float x2 = qpl[n * 144 + 96 + k];
    int h = k >> 2, p = k & 3;
#pragma unroll
    for (int a = 0; a < 3; ++a) {
      float g = R[a * 3 + 0] * x0 + R[a * 3 + 1] * x1 + R[a * 3 + 2] * x2 + t[a];
      qpt[((size_t)(a * HV + h) * N_RES + n) * PQKV + p] = g;
      sqh[h] += g * g;
    }
  }
  // key/value points: kvpl is [n][3][H*(PQK+PV)]
  for (int k = 0; k < HV * (PQKV + PVV); ++k) {
    float x0 = kvpl[n * 432 + k];
    float x1 = kvpl[n * 432 + 144 + k];
    float x2 = kvpl[n * 432 + 288 + k];
    int h = k / 12, j = k % 12;
#pragma unroll
    for (int a = 0; a < 3; ++a) {
      float g = R[a * 3 + 0] * x0 + R[a * 3 + 1] * x1 + R[a * 3 + 2] * x2 + t[a];
      if (j < PQKV) {
        kpt[((size_t)(a * HV + h) * N_RES + n) * PQKV + j] = g;
        skh[h] += g * g;
      } else {
        vpt[((size_t)(a * HV + h) * N_RES + n) * PVV + (j - PQKV)] = g;
      }
    }
  }
#pragma unroll
  for (int h = 0; h < HV; ++h) {
    sq[h * N_RES + n] = sqh[h];
    sk[h * N_RES + n] = skh[h];
  }
}

// ---------------------------------------------------------------------------
// logits[h][n][m] = scalar_w * (q.k) - 0.5*point_w[h]*(sq+sk-2*qp.kp)
// One wave per (h, n16, m16) tile: 4 wmma (scalar, K=16) + 3 wmma (points).
// ---------------------------------------------------------------------------
__global__ void logits_qk_kernel(const float* __restrict__ qs,
                                 const float* __restrict__ kvs,
                                 const float* __restrict__ qpt,
                                 const float* __restrict__ kpt,
                                 const float* __restrict__ sq,
                                 const float* __restrict__ sk,
                                 const float* __restrict__ tpw,
                                 float* __restrict__ logits) {
  const int TN = N_RES >> 4;
  int w = wave_linear_id();
  if (w >= HV * TN * TN) return;
  int mt = w % TN, nt = (w / TN) % TN, h = w / (TN * TN);
  int n0 = nt << 4, m0 = mt << 4;
  int l = threadIdx.x & 31, half = l >> 4, lm = l & 15;

  v8f accS = {0.f, 0.f, 0.f, 0.f, 0.f, 0.f, 0.f, 0.f};
  v8f accP = {0.f, 0.f, 0.f, 0.f, 0.f, 0.f, 0.f, 0.f};

  // scalar QK: A row n, B col m (k_scalar stored m-major so loads mirror A)
  const float* qrow = qs + (size_t)(n0 + lm) * (HV * SQKV) + h * SQKV + 2 * half;
  const float* krow = kvs + (size_t)(m0 + lm) * (HV * 32) + h * 32 + 2 * half;
#pragma unroll
  for (int kk = 0; kk < SQKV; kk += 4) {
    v2f a, b;
    a.x = qrow[kk]; a.y = qrow[kk + 1];
    b.x = krow[kk]; b.y = krow[kk + 1];
    accS = wmma_f32_k4(a, b, accS);
  }
  // point QK: 3 coords x K=PQK(=4) each -> one wmma per coordinate
#pragma unroll
  for (int a3 = 0; a3 < 3; ++a3) {
    const float* qp = qpt + ((size_t)(a3 * HV + h) * N_RES + n0 + lm) * PQKV + 2 * half;
    const float* kp = kpt + ((size_t)(a3 * HV + h) * N_RES + m0 + lm) * PQKV + 2 * half;
    v2f a, b;
    a.x = qp[0]; a.y = qp[1];
    b.x = kp[0]; b.y = kp[1];
    accP = wmma_f32_k4(a, b, accP);
  }

  float pw = sqrtf(1.0f / 54.0f) * log1pf(expf(tpw[h]));  // 3*PQK*9/2 = 54
  const float scalar_w = sqrtf(1.0f / 48.0f);             // 3*SQK = 48
#pragma unroll
  for (int r = 0; r < 8; ++r) {
    int n = n0 + r + 8 * half;
    int m = m0 + lm;
    float d2 = sq[h * N_RES + n] + sk[h * N_RES + m] - 2.0f * accP[r];
    logits[((size_t)h * N_RES + n) * N_RES + m] = scalar_w * accS[r] - 0.5f * pw * d2;
  }
}

// ---------------------------------------------------------------------------
// logits[h][n][m] += sqrt(1/3) * (inputs_2d[n][m][:] @ w2d[:,h] + b2d[h])
// One wave per (n, m16) tile: M=m rows, N=heads (12 of 16), K=128.
// Streams inputs_2d exactly once. Padded B-cols (lm>=12) read col 0; the
// garbage lands only in D-cols we never store.
// ---------------------------------------------------------------------------
__global__ void att2d_add_kernel(const float* __restrict__ in2d,
                                 const float* __restrict__ w2d,
                                 const float* __restrict__ b2d,
                                 float* __restrict__ logits) {
  const int TM = N_RES >> 4;
  int w = wave_linear_id();
  if (w >= N_RES * TM) return;
  int mt = w % TM, n = w / TM;
  int m0 = mt << 4;
  int l = threadIdx.x & 31, half = l >> 4, lm = l & 15;
  int lmh = (lm < HV) ? lm : 0;  // clamp once; padded cols are dead

  v8f acc = {0.f, 0.f, 0.f, 0.f, 0.f, 0.f, 0.f, 0.f};
  const float* arow = in2d + ((size_t)n * N_RES + m0 + lm) * C2V + 2 * half;
  const float* bcol = w2d + 2 * half * HV + lmh;
  // prefetch the next m-tile's A rows (same n slab, +16 rows ahead)
  __builtin_prefetch(arow + (size_t)16 * C2V, 0, 1);
#pragma unroll 8
  for (int kk = 0; kk < C2V; kk += 4) {
    v2f a, b;
    a.x = arow[kk];
    a.y = arow[kk + 1];
    b.x = bcol[kk * HV];
    b.y = bcol[(kk + 1) * HV];
    acc = wmma_f32_k4(a, b, acc);
  }
  const float c3 = sqrtf(1.0f / 3.0f);
  if (lm < HV) {
    float bb = b2d[lm];
#pragma unroll
    for (int r = 0; r < 8; ++r) {
      int m = m0 + r + 8 * half;
      logits[((size_t)lm * N_RES + n) * N_RES + m] += c3 * (acc[r] + bb);
    }
  }
}

// ---------------------------------------------------------------------------
// Row softmax with mask: one 256-thread block per (h,n) row of 1024.
// ---------------------------------------------------------------------------
__global__ void softmax_kernel(float* __restrict__ logits,
                               const float* __restrict__ mask) {
  int row = blockIdx.x;  // h*N + n
  int n = row & (N_RES - 1);
  float* p = logits + (size_t)row * N_RES;
  int tid = threadIdx.x;
  float mn = mask[n];

  float v[4];
  float mx = -3.4e38f;
#pragma unroll
  for (int i = 0; i < 4; ++i) {
    int m = tid + i * 256;
    v[i] = p[m] - 100000.0f * (1.0f - mn * mask[m]);
    mx = fmaxf(mx, v[i]);
  }
  __shared__ float red[256];
  red[tid] = mx;
  __syncthreads();
  for (int s = 128; s > 0; s >>= 1) {
    if (tid < s) red[tid] = fmaxf(red[tid], red[tid + s]);
    __syncthreads();
  }
  mx = red[0];
  __syncthreads();
  float sum = 0.f;
#pragma unroll
  for (int i = 0; i < 4; ++i) { v[i] = expf(v[i] - mx); sum += v[i]; }
  red[tid] = sum;
  __syncthreads();
  for (int s = 128; s > 0; s >>= 1) {
    if (tid < s) red[tid] += red[tid + s];
    __syncthreads();
  }
  float inv = 1.0f / red[0];
#pragma unroll
  for (int i = 0; i < 4; ++i) p[tid + i * 256] = v[i] * inv;
}

// ---------------------------------------------------------------------------
// res_scalar[n][h][s] = sum_m attn[h][n][m] * v_scalar[m][h][s]  (K=1024)
// writes into final[:, h*16 + s]
// ---------------------------------------------------------------------------
__global__ void res_scalar_kernel(const float* __restrict__ attn,
                                  const float* __restrict__ kvs,
                                  float* __restrict__ fb) {
  const int TN = N_RES >> 4;
  int w = wave_linear_id();
  if (w >= HV * TN) return;
  int nt = w % TN, h = w / TN;
  int n0 = nt << 4;
  int l = threadIdx.x & 31, half = l >> 4, lm = l & 15;

  v8f acc = {0.f, 0.f, 0.f, 0.f, 0.f, 0.f, 0.f, 0.f};
  const float* arow = attn + ((size_t)h * N_RES + n0 + lm) * N_RES + 2 * half;
  const float* bcol = kvs + (size_t)2 * half * (HV * 32) + h * 32 + SQKV + lm;
#pragma unroll 8
  for (int kk = 0; kk < N_RES; kk += 4) {
    v2f a, b;
    a.x = arow[kk];
    a.y = arow[kk + 1];
    b.x = bcol[(size_t)kk * (HV * 32)];
    b.y = bcol[(size_t)(kk + 1) * (HV * 32)];
    acc = wmma_f32_k4(a, b, acc);
  }
#pragma unroll
  for (int r = 0; r < 8; ++r) {
    int n = n0 + r + 8 * half;
    fb[(size_t)n * DOUTV + h * SVV + lm] = acc[r];
  }
}

// ---------------------------------------------------------------------------
// res_pg[c][n][h*PV+p] = sum_m attn[h][n][m] * v_pt[c][h][m][p]
// Padded B-cols (lm>=8) alias col lm&7; their D-cols are never stored.
// ---------------------------------------------------------------------------
__global__ void res_point_kernel(const float* __restrict__ attn,
                                 const float* __restrict__ vpt,
                                 float* __restrict__ rpg) {
  const int TN = N_RES >> 4;
  int w = wave_linear_id();
  if (w >= HV * 3 * TN) return;
  int nt = w % TN, c = (w / TN) % 3, h = w / (TN * 3);
  int n0 = nt << 4;
  int l = threadIdx.x & 31, half = l >> 4, lm = l & 15;
  int lmp = lm & (PVV - 1);  // clamp once; padded cols are dead

  v8f acc = {0.f, 0.f, 0.f, 0.f, 0.f, 0.f, 0.f, 0.f};
  const float* arow = attn + ((size_t)h * N_RES + n0 + lm) * N_RES + 2 * half;
  const float* bcol =
      vpt + ((size_t)(c * HV + h) * N_RES + 2 * half) * PVV + lmp;
#pragma unroll 8
  for (int kk = 0; kk < N_RES; kk += 4) {
    v2f a, b;
    a.x = arow[kk];
    a.y = arow[kk + 1];
    b.x = bcol[(size_t)kk * PVV];
    b.y = bcol[(size_t)(kk + 1) * PVV];
    acc = wmma_f32_k4(a, b, acc);
  }
  if (lm < PVV) {
#pragma unroll
    for (int r = 0; r < 8; ++r) {
      int n = n0 + r + 8 * half;
      rpg[((size_t)c * N_RES + n) * (HV * PVV) + h * PVV + lm] = acc[r];
    }
  }
}

// ---------------------------------------------------------------------------
// res_2d[n][h][c] = sum_m attn[h][n][m] * inputs_2d[n][m][c]
// One wave per (n, c16): M = heads (12 of 16), K = 1024. Second streaming
// pass over inputs_2d. Padded A-rows (lm>=12) alias row 0; their D-rows are
// never stored. 8 c-tiles of one n share a block for L2 locality.
// ---------------------------------------------------------------------------
__global__ void res2d_kernel(const float* __restrict__ attn,
                             const float* __restrict__ in2d,
                             float* __restrict__ fb) {
  const int TC = C2V >> 4;  // 8
  int w = wave_linear_id();
  if (w >= N_RES * TC) return;
  int ct = w % TC, n = w / TC;
  int c0 = ct << 4;
  int l = threadIdx.x & 31, half = l >> 4, lm = l & 15;
  int lmh = (lm < HV) ? lm : 0;  // clamp once; padded rows are dead

  v8f acc = {0.f, 0.f, 0.f, 0.f, 0.f, 0.f, 0.f, 0.f};
  const float* arow = attn + ((size_t)lmh * N_RES + n) * N_RES + 2 * half;
  const float* brow = in2d + ((size_t)n * N_RES + 2 * half) * C2V + c0 + lm;
#pragma unroll 8
  for (int kk = 0; kk < N_RES; kk += 4) {
    v2f a, b;
    a.x = arow[kk];
    a.y = arow[kk + 1];
    b.x = brow[(size_t)kk * C2V];
    b.y = brow[(size_t)(kk + 1) * C2V];
    // prefetch B 32 K-rows (16 KB) ahead in the n-slab stream
    __builtin_prefetch(brow + (size_t)(kk + 32) * C2V, 0, 1);
    acc = wmma_f32_k4(a, b, acc);
  }
#pragma unroll
  for (int r = 0; r < 8; ++r) {
    int h = r + 8 * half;
    if (h < HV)
      fb[(size_t)n * DOUTV + 576 + h * C2V + c0 + lm] = acc[r];
  }
}

// ---------------------------------------------------------------------------
// res_pl[a][n][k] = sum_c rot[n][c][a] * (res_pg[c][n][k] - trans[n][c])
// pt_norm[n][k] = sqrt(eps + sum_a res_pl^2). Fills final[192..576).
// ---------------------------------------------------------------------------
__global__ void respl_norm_kernel(const float* __restrict__ rpg,
                                  const float* __restrict__ rot,
                                  const float* __restrict__ trans,
                                  float* __restrict__ fb) {
  int idx = blockIdx.x * blockDim.x + threadIdx.x;
  if (idx >= N_RES * HV * PVV) return;
  int n = idx / 96, k = idx % 96;
  float x[3];
#pragma unroll
  for (int c = 0; c < 3; ++c)
    x[c] = rpg[((size_t)c * N_RES + n) * 96 + k] - trans[n * 3 + c];
  float s = 0.f;
#pragma unroll
  for (int a = 0; a < 3; ++a) {
    float r = rot[n * 9 + 0 + a] * x[0] + rot[n * 9 + 3 + a] * x[1] +
              rot[n * 9 + 6 + a] * x[2];
    fb[(size_t)n * DOUTV + 192 + a * 96 + k] = r;
    s += r * r;
  }
  fb[(size_t)n * DOUTV + 480 + k] = sqrtf(1e-8f + s);
}

// ---------------------------------------------------------------------------
extern "C" void kernel_launch(void* const* d_in, const int* in_sizes, int n_in,
                              void* d_out, int out_size, void* d_ws,
                              size_t ws_size, hipStream_t stream) {
  const float* inputs_1d = (const float*)d_in[0];
  const float* inputs_2d = (const float*)d_in[1];
  const float* mask      = (const float*)d_in[2];
  const float* rot       = (const float*)d_in[3];
  const float* trans     = (const float*)d_in[4];
  const float* wq        = (const float*)d_in[5];
  const float* bq        = (const float*)d_in[6];
  const float* wkv       = (const float*)d_in[7];
  const float* bkv       = (const float*)d_in[8];
  const float* wqp       = (const float*)d_in[9];
  const float* bqp       = (const float*)d_in[10];
  const float* wkvp      = (const float*)d_in[11];
  const float* bkvp      = (const float*)d_in[12];
  const float* w2d       = (const float*)d_in[13];
  const float* b2d       = (const float*)d_in[14];
  const float* tpw       = (const float*)d_in[15];
  const float* wo        = (const float*)d_in[16];
  const float* bo        = (const float*)d_in[17];
  float* out = (float*)d_out;

  float* p = (float*)d_ws;
  float* q_scalar = p;  p += (size_t)N_RES * 192;
  float* kv_scalar = p; p += (size_t)N_RES * 384;
  float* q_pl = p;      p += (size_t)N_RES * 144;
  float* kv_pl = p;     p += (size_t)N_RES * 432;
  float* qpt = p;       p += (size_t)3 * HV * N_RES * PQKV;
  float* kpt = p;       p += (size_t)3 * HV * N_RES * PQKV;
  float* vpt = p;       p += (size_t)3 * HV * N_RES * PVV;
  float* sq = p;        p += (size_t)HV * N_RES;
  float* sk = p;        p += (size_t)HV * N_RES;
  float* rpg = p;       p += (size_t)3 * N_RES * (HV * PVV);
  float* fb = p;        p += (size_t)N_RES * DOUTV;
  float* logits = p;    p += (size_t)HV * N_RES * N_RES;

  auto waves_grid = [](long long waves) { return (int)((waves + 7) / 8); };
  dim3 blk(256);

  // 1) projections (fp32 WMMA GEMMs)
  gemm_wmma_f32<<<waves_grid(64LL * 12), blk, 0, stream>>>(
      inputs_1d, wq, bq, q_scalar, N_RES, C1V, 192);
  gemm_wmma_f32<<<waves_grid(64LL * 24), blk, 0, stream>>>(
      inputs_1d, wkv, bkv, kv_scalar, N_RES, C1V, 384);
  gemm_wmma_f32<<<waves_grid(64LL * 9), blk, 0, stream>>>(
      inputs_1d, wqp, bqp, q_pl, N_RES, C1V, 144);
  gemm_wmma_f32<<<waves_grid(64LL * 27), blk, 0, stream>>>(
      inputs_1d, wkvp, bkvp, kv_pl, N_RES, C1V, 432);

  // 2) rigid-frame transform + norms
  rotate_points_kernel<<<dim3(N_RES / 256), blk, 0, stream>>>(
      q_pl, kv_pl, rot, trans, qpt, kpt, vpt, sq, sk);

  // 3) QK logits (scalar + point) via WMMA
  logits_qk_kernel<<<waves_grid((long long)HV * 64 * 64), blk, 0, stream>>>(
      q_scalar, kv_scalar, qpt, kpt, sq, sk, tpw, logits);

  // 4) pair-bias term: stream inputs_2d once through WMMA
  att2d_add_kernel<<<waves_grid((long long)N_RES * 64), blk, 0, stream>>>(
      inputs_2d, w2d, b2d, logits);

  // 5) masked softmax in place
  softmax_kernel<<<dim3(HV * N_RES), blk, 0, stream>>>(logits, mask);

  // 6) attn @ v_scalar
  res_scalar_kernel<<<waves_grid((long long)HV * 64), blk, 0, stream>>>(
      logits, kv_scalar, fb);

  // 7) attn @ v_pt
  res_point_kernel<<<waves_grid((long long)HV * 3 * 64), blk, 0, stream>>>(
      logits, vpt, rpg);

  // 8) attn @ inputs_2d (second streaming pass)
  res2d_kernel<<<waves_grid((long long)N_RES * 8), blk, 0, stream>>>(
      logits, inputs_2d, fb);

  // 9) back to local frame + point norms
  respl_norm_kernel<<<dim3((N_RES * 96) / 256), blk, 0, stream>>>(
      rpg, rot, trans, fb);

  // 10) output projection: final @ wo + bo
  gemm_wmma_f32<<<waves_grid(64LL * 24), blk, 0, stream>>>(
      fb, wo, bo, out, N_RES, DOUTV, OUTV);
}